// GNG_ODE_7172595384550
// MI455X (gfx1250) — compile-verified
//
#include <hip/hip_runtime.h>
#include <hip/hip_bf16.h>
#include <math.h>
#include <stdint.h>

// ---------------------------------------------------------------------------
// CDNA5 / gfx1250: wave32; V_WMMA_F32_16X16X4_F32 for all GEMMs;
// GLOBAL_LOAD_ASYNC_TO_LDS_B128 (+ s_wait_asynccnt) to stage the 51 MB target
// matrix through LDS for the dominant 512 x 100000 x 128 logits GEMM.
// ---------------------------------------------------------------------------

#ifndef __has_builtin
#define __has_builtin(x) 0
#endif

typedef __attribute__((ext_vector_type(2))) float v2f;
typedef __attribute__((ext_vector_type(8))) float v8f;
typedef __attribute__((__vector_size__(16))) int v4i;   // builtin's 16B unit

constexpr int kB   = 512;
constexpr int kNPG = 50;
constexpr int kEPG = 60;
constexpr int kD   = 128;
constexpr int kV   = 100000;
constexpr int kN   = kB * kNPG;   // 25600
constexpr int kE   = kB * kEPG;   // 30720
constexpr int kSplits = 4;
constexpr float kScale = 12.0f;
constexpr int WPB = 8;            // waves (wave32) per 256-thread block

// ---------------- wave reductions (wave32!) ----------------
__device__ inline float waveSum(float v) {
#pragma unroll
  for (int off = 16; off; off >>= 1) v += __shfl_xor(v, off, 32);
  return v;
}
__device__ inline float waveMax(float v) {
#pragma unroll
  for (int off = 16; off; off >>= 1) v = fmaxf(v, __shfl_xor(v, off, 32));
  return v;
}
__device__ inline float sigmoidf(float x) { return 1.0f / (1.0f + expf(-x)); }

// ---------------- CDNA5 async global->LDS helpers ----------------
// lds_byte: LDS byte offset (AS(3) pointers are the raw LDS byte offset on
// AMDGPU; low 32 bits of a generic shared pointer are that offset per the
// gfx1250 aperture mapping). 16 bytes per call; tracked on ASYNCcnt.
__device__ __forceinline__ void async_copy_b128(uint32_t lds_byte, const void* gptr) {
#if __has_builtin(__builtin_amdgcn_global_load_async_to_lds_b128)
  typedef __attribute__((address_space(1))) v4i* gp_t;
  typedef __attribute__((address_space(3))) v4i* lp_t;
  __builtin_amdgcn_global_load_async_to_lds_b128(
      (gp_t)(uintptr_t)gptr, (lp_t)(unsigned)lds_byte, 0, 0);
#else
  asm volatile("global_load_async_to_lds_b128 %0, %1, off"
               : : "v"(lds_byte), "v"(gptr) : "memory");
#endif
}
__device__ __forceinline__ void wait_async0() {
#if __has_builtin(__builtin_amdgcn_s_wait_asynccnt)
  __builtin_amdgcn_s_wait_asynccnt(0);
#else
  asm volatile("s_wait_asynccnt 0x0" : : : "memory");
#endif
}

// ---------------- utility kernels ----------------
__global__ void k_fill(float* __restrict__ p, float v, size_t n) {
  size_t i  = (size_t)blockIdx.x * blockDim.x + threadIdx.x;
  size_t st = (size_t)gridDim.x * blockDim.x;
  for (; i < n; i += st) p[i] = v;
}

// out[row] = in[row] / max(||in[row]||, 1e-12); one wave per 128-float row
__global__ void k_l2norm_rows(const float* __restrict__ in, float* __restrict__ out, int rows) {
  int row = blockIdx.x * (blockDim.x >> 5) + (threadIdx.x >> 5);
  if (row >= rows) return;
  int lane = threadIdx.x & 31;
  float4 v = ((const float4*)(in + (size_t)row * kD))[lane];
  float ss = waveSum(v.x * v.x + v.y * v.y + v.z * v.z + v.w * v.w);
  float inv = 1.0f / fmaxf(sqrtf(ss), 1e-12f);
  float4 o = make_float4(v.x * inv, v.y * inv, v.z * inv, v.w * inv);
  ((float4*)(out + (size_t)row * kD))[lane] = o;
}

// out[row] = tbl[idx[row]]; one wave per row
__global__ void k_gather_rows(const float* __restrict__ tbl, const int* __restrict__ idx,
                              float* __restrict__ out, int rows) {
  int row = blockIdx.x * (blockDim.x >> 5) + (threadIdx.x >> 5);
  if (row >= rows) return;
  int lane = threadIdx.x & 31;
  int v = idx[row];
  ((float4*)(out + (size_t)row * kD))[lane] = ((const float4*)(tbl + (size_t)v * kD))[lane];
}

// positive-float max reduce via int atomicMax on the bit pattern
__global__ void k_reduce_max_pos(const float* __restrict__ v, int n, float* __restrict__ out) {
  float m = 0.0f;
  for (int i = blockIdx.x * blockDim.x + threadIdx.x; i < n; i += gridDim.x * blockDim.x)
    m = fmaxf(m, v[i]);
  m = waveMax(m);
  if ((threadIdx.x & 31) == 0) atomicMax((int*)out, __float_as_int(m));
}

// ---------------- generic WMMA f32 GEMM ----------------
// C[M x Nc] = alpha * (diag(rowScale) * A) @ B (+ bias[col]) (+ C if accumulate)
// A row-major (lda). B addressed as B[k*ldbK + col*ldbN] -> transB is free.
// One 16x16 output tile per wave; K-loop in steps of 4 (V_WMMA_F32_16X16X4_F32).
__global__ void k_wmma_gemm(const float* __restrict__ A, int lda,
                            const float* __restrict__ B, int ldbK, int ldbN,
                            float* __restrict__ C, int ldc,
                            const float* __restrict__ bias,
                            const float* __restrict__ rowScale,
                            int M, int Nc, int K, float alpha, int accumulate) {
  int wave   = blockIdx.x * (blockDim.x >> 5) + (threadIdx.x >> 5);
  int tilesN = Nc >> 4;
  int tiles  = (M >> 4) * tilesN;
  if (wave >= tiles) return;            // uniform per-wave: EXEC all-ones inside
  int lane = threadIdx.x & 31;
  int tm = wave / tilesN;
  int tn = wave - tm * tilesN;
  int row0 = tm << 4, col0 = tn << 4;
  int half = lane >> 4, r = lane & 15;

  float s = 1.0f;
  if (rowScale) s = rowScale[row0 + r];
  const float* arow = A + (size_t)(row0 + r) * lda;
  const float* bcol = B + (size_t)(col0 + r) * ldbN;

  v8f acc = {0.f, 0.f, 0.f, 0.f, 0.f, 0.f, 0.f, 0.f};
  for (int k0 = 0; k0 < K; k0 += 4) {
    int ka = k0 + (half << 1);
    v2f a, b;
    a.x = arow[ka] * s;
    a.y = arow[ka + 1] * s;
    b.x = bcol[(size_t)ka * ldbK];
    b.y = bcol[(size_t)(ka + 1) * ldbK];
    acc = __builtin_amdgcn_wmma_f32_16x16x4_f32(false, a, false, b, (short)0, acc,
                                                false, false);
  }
  float bc = bias ? bias[col0 + r] : 0.0f;
#pragma unroll
  for (int i = 0; i < 8; ++i) {
    int row = row0 + i + (half << 3);
    size_t o = (size_t)row * ldc + (size_t)(col0 + r);
    float v = alpha * acc[i] + bc;
    if (accumulate) v += C[o];
    C[o] = v;
  }
}

// ---------------- logits GEMM: C = SCALE * A(512x128) @ target^T ----------------
// Per block: stage 128 contiguous target rows (64 KB) into LDS with async
// B128 transfers; 8 waves each compute one 16x16 tile, B frags via ds_load.
constexpr int LCOLS = 128;  // target rows (= logit cols) per block
__global__ void __launch_bounds__(256)
k_logits_gemm(const float* __restrict__ A, const float* __restrict__ Bt,
              float* __restrict__ C) {
  __shared__ float smem[LCOLS * kD];   // 64 KB
  int vbase = blockIdx.x * LCOLS;
  int row0  = blockIdx.y * 16;
  int tid   = threadIdx.x;

  // ---- async stage: 64 KB contiguous (rows vbase..vbase+127 of Bt) ----
  const float* gsrc = Bt + (size_t)vbase * kD;
  uint32_t ldsbase = (uint32_t)(uintptr_t)smem;  // LDS byte offset
#pragma unroll
  for (int i = 0; i < 16; ++i) {
    int f = (tid + i * 256) * 4;                 // float index, 16B chunks
    async_copy_b128(ldsbase + (uint32_t)f * 4u, gsrc + f);
  }
  wait_async0();
  __syncthreads();

  int wid = tid >> 5, lane = tid & 31;
  int half = lane >> 4, r = lane & 15;
  int colL = wid * 16;                           // this wave's local col tile
  const float* arow = A + (size_t)(row0 + r) * kD;

  v8f acc = {0.f, 0.f, 0.f, 0.f, 0.f, 0.f, 0.f, 0.f};
  for (int k0 = 0; k0 < kD; k0 += 4) {
    int ka = k0 + (half << 1);
    v2f a, b;
    a.x = arow[ka];
    a.y = arow[ka + 1];
    b.x = smem[(colL + r) * kD + ka];            // ds_load (consecutive -> b64)
    b.y = smem[(colL + r) * kD + ka + 1];
    acc = __builtin_amdgcn_wmma_f32_16x16x4_f32(false, a, false, b, (short)0, acc,
                                                false, false);
  }
  int col0 = vbase + colL;
  if (col0 < kV) {                               // wave-uniform tail guard
#pragma unroll
    for (int i = 0; i < 8; ++i) {
      int row = row0 + i + (half << 3);
      C[(size_t)row * kV + (size_t)(col0 + r)] = kScale * acc[i];
    }
  }
}

// ---------------- GRU-phase edge aggregation ----------------
__global__ void k_gru_edge_scatter(const int* __restrict__ src, const int* __restrict__ dst,
                                   const float* __restrict__ ew, const float* __restrict__ feat,
                                   float* __restrict__ num1, float* __restrict__ num2,
                                   float* __restrict__ den1, float* __restrict__ den2) {
  int e = blockIdx.x * (blockDim.x >> 5) + (threadIdx.x >> 5);
  if (e >= kE) return;
  int lane = threadIdx.x & 31;
  int s = src[e], d = dst[e];
  float w = ew[e];
  int c0 = lane * 4;
  const float* fs = feat + (size_t)s * kD + c0;
  const float* fd = feat + (size_t)d * kD + c0;
  float* n1 = num1 + (size_t)d * kD + c0;
  float* n2 = num2 + (size_t)s * kD + c0;
#pragma unroll
  for (int i = 0; i < 4; ++i) {
    atomicAdd(n1 + i, fs[i] * w);
    atomicAdd(n2 + i, fd[i] * w);
  }
  if (lane == 0) { atomicAdd(den1 + d, w); atomicAdd(den2 + s, w); }
}

// num[i] /= (den[row]==0 ? 1 : den[row])
__global__ void k_div_den(float* __restrict__ num, const float* __restrict__ den, size_t total) {
  size_t i  = (size_t)blockIdx.x * blockDim.x + threadIdx.x;
  size_t st = (size_t)gridDim.x * blockDim.x;
  for (; i < total; i += st) {
    float d = den[i >> 7];
    num[i] /= (d == 0.0f ? 1.0f : d);
  }
}

// GRU gates + l2norm, fused; writes both x (feat) and h (ODE init)
__global__ void k_gru_gate_norm(const float* __restrict__ gi, const float* __restrict__ gh,
                                float* __restrict__ x, float* __restrict__ h, int rows) {
  int row = blockIdx.x * (blockDim.x >> 5) + (threadIdx.x >> 5);
  if (row >= rows) return;
  int lane = threadIdx.x & 31;
  const float* gir = gi + (size_t)row * 384;
  const float* ghr = gh + (size_t)row * 384;
  float* xr = x + (size_t)row * kD;
  float* hr = h + (size_t)row * kD;
  float vals[4];
  float ss = 0.0f;
#pragma unroll
  for (int i = 0; i < 4; ++i) {
    int d = lane * 4 + i;
    float r_ = sigmoidf(gir[d] + ghr[d]);
    float z_ = sigmoidf(gir[128 + d] + ghr[128 + d]);
    float g_ = tanhf(gir[256 + d] + r_ * ghr[256 + d]);
    float fn = (1.0f - z_) * g_ + z_ * xr[d];
    vals[i] = fn;
    ss += fn * fn;
  }
  ss = waveSum(ss);
  float inv = 1.0f / fmaxf(sqrtf(ss), 1e-12f);
#pragma unroll
  for (int i = 0; i < 4; ++i) {
    int d = lane * 4 + i;
    float v = vals[i] * inv;
    xr[d] = v;
    hr[d] = v;
  }
}

// ---------------- ODE kernels ----------------
__global__ void k_ode_mask_deg(const int* __restrict__ src, const int* __restrict__ dst,
                               const float* __restrict__ et, const float* __restrict__ nt,
                               const float* __restrict__ tmax, int kstep,
                               float* __restrict__ emf, float* __restrict__ deg) {
  int e = blockIdx.x * blockDim.x + threadIdx.x;
  if (e >= kE) return;
  float t = (*tmax) * ((float)kstep / (float)kSplits);
  int s = src[e], d = dst[e];
  float m = (et[e] <= t && nt[s] >= t && nt[d] >= t && s != d) ? 1.0f : 0.0f;
  emf[e] = m;
  if (m != 0.0f) { atomicAdd(deg + s, 1.0f); atomicAdd(deg + d, 1.0f); }
}

__global__ void k_compute_dis(const float* __restrict__ deg, float* __restrict__ dis, int n) {
  int i = blockIdx.x * blockDim.x + threadIdx.x;
  if (i < n) dis[i] = rsqrtf(fmaxf(deg[i], 1.0f));  // clip(deg,1)^-0.5
}

__global__ void k_ode_scatter_xh(const int* __restrict__ src, const int* __restrict__ dst,
                                 const float* __restrict__ emf, const float* __restrict__ dis,
                                 const float* __restrict__ X, const float* __restrict__ H,
                                 float* __restrict__ aggX, float* __restrict__ aggH) {
  int e = blockIdx.x * (blockDim.x >> 5) + (threadIdx.x >> 5);
  if (e >= kE) return;
  if (emf[e] == 0.0f) return;     // uniform per-wave
  int lane = threadIdx.x & 31;
  int s = src[e], d = dst[e];
  float ds_ = dis[s], dd = dis[d];
  int c0 = lane * 4;
  const float* xs = X + (size_t)s * kD + c0;
  const float* xd = X + (size_t)d * kD + c0;
  const float* hs = H + (size_t)s * kD + c0;
  const float* hd = H + (size_t)d * kD + c0;
  float* axd = aggX + (size_t)d * kD + c0;
  float* axs = aggX + (size_t)s * kD + c0;
  float* ahd = aggH + (size_t)d * kD + c0;
  float* ahs = aggH + (size_t)s * kD + c0;
#pragma unroll
  for (int i = 0; i < 4; ++i) {
    atomicAdd(axd + i, xs[i] * ds_);
    atomicAdd(axs + i, xd[i] * dd);
    atomicAdd(ahd + i, hs[i] * ds_);
    atomicAdd(ahs + i, hd[i] * dd);
  }
}

__global__ void k_ode_scatter_one(const int* __restrict__ src, const int* __restrict__ dst,
                                  const float* __restrict__ emf, const float* __restrict__ dis,
                                  const float* __restrict__ X, float* __restrict__ agg) {
  int e = blockIdx.x * (blockDim.x >> 5) + (threadIdx.x >> 5);
  if (e >= kE) return;
  if (emf[e] == 0.0f) return;
  int lane = threadIdx.x & 31;
  int s = src[e], d = dst[e];
  float ds_ = dis[s], dd = dis[d];
  int c0 = lane * 4;
  const float* xs = X + (size_t)s * kD + c0;
  const float* xd = X + (size_t)d * kD + c0;
  float* ad = agg + (size_t)d * kD + c0;
  float* as = agg + (size_t)s * kD + c0;
#pragma unroll
  for (int i = 0; i < 4; ++i) {
    atomicAdd(ad + i, xs[i] * ds_);
    atomicAdd(as + i, xd[i] * dd);
  }
}

// rbuf -> sigmoid(rbuf); zbuf -> sigmoid(zbuf); rh = rr * h
__global__ void k_ode_gates_rz(float* __restrict__ rbuf, float* __restrict__ zbuf,
                               const float* __restrict__ h, float* __restrict__ rh,
                               size_t total) {
  size_t i  = (size_t)blockIdx.x * blockDim.x + threadIdx.x;
  size_t st = (size_t)gridDim.x * blockDim.x;
  for (; i < total; i += st) {
    float r = sigmoidf(rbuf[i]);
    float z = sigmoidf(zbuf[i]);
    rbuf[i] = r;
    zbuf[i] = z;
    rh[i] = r * h[i];
  }
}

// h += dt * where(node_t>=t, (1-z)*(tanh(u)-h), 0)
__global__ void k_ode_update_h(const float* __restrict__ ubuf, const float* __restrict__ zbuf,
                               float* __restrict__ h, const float* __restrict__ nt,
                               const float* __restrict__ tmax, int kstep, size_t total) {
  size_t i  = (size_t)blockIdx.x * blockDim.x + threadIdx.x;
  size_t st = (size_t)gridDim.x * blockDim.x;
  float dt = (*tmax) / (float)kSplits;
  float t = dt * (float)kstep;
  for (; i < total; i += st) {
    int row = (int)(i >> 7);
    if (nt[row] >= t) {
      float u = tanhf(ubuf[i]);
      float z = zbuf[i];
      float hh = h[i];
      h[i] = hh + dt * ((1.0f - z) * (u - hh));
    }
  }
}

// ---------------- attention readout ----------------
__global__ void k_attn_e(const float* __restrict__ uall, const float* __restrict__ lv,
                         const float* __restrict__ We, float* __restrict__ evec, int rows) {
  int row = blockIdx.x * (blockDim.x >> 5) + (threadIdx.x >> 5);
  if (row >= rows) return;
  int lane = threadIdx.x & 31;
  int b = row / kNPG;
  const float* ur = uall + (size_t)row * kD;
  const float* lr = lv + (size_t)b * kD;
  float acc = 0.0f;
#pragma unroll
  for (int i = 0; i < 4; ++i) {
    int d = lane * 4 + i;
    acc += sigmoidf(ur[d] + lr[d]) * We[d];
  }
  acc = waveSum(acc);
  if (lane == 0) evec[row] = acc;
}

// per-session softmax over NPG + weighted sum; builds srin = [last, sr_g]
__global__ void k_attn_combine(const float* __restrict__ fb, const float* __restrict__ evec,
                               float* __restrict__ srin) {
  int b = blockIdx.x;        // 512 blocks, 128 threads
  int d = threadIdx.x;
  __shared__ float al[kNPG];
  if (d < kNPG) al[d] = evec[b * kNPG + d];
  __syncthreads();
  if (d == 0) {
    float m = -3.4e38f;
    for (int j = 0; j < kNPG; ++j) m = fmaxf(m, al[j]);
    float s = 0.0f;
    for (int j = 0; j < kNPG; ++j) { al[j] = expf(al[j] - m); s += al[j]; }
    float inv = 1.0f / s;
    for (int j = 0; j < kNPG; ++j) al[j] *= inv;
  }
  __syncthreads();
  const float* base = fb + ((size_t)b * kNPG) * kD;
  float acc = 0.0f;
  for (int j = 0; j < kNPG; ++j) acc += al[j] * base[(size_t)j * kD + d];
  srin[(size_t)b * 256 + d] = base[(size_t)(kNPG - 1) * kD + d];  // last
  srin[(size_t)b * 256 + 128 + d] = acc;                          // sr_g
}

// ---------------- row log-softmax over V ----------------
__global__ void k_log_softmax_rows(float* __restrict__ out, int cols) {
  int b = blockIdx.x;
  float* row = out + (size_t)b * cols;
  __shared__ float sred[8];
  int tid = threadIdx.x, lane = tid & 31, wid = tid >> 5, nw = blockDim.x >> 5;

  float m = -3.4e38f;
  for (int i = tid; i < cols; i += blockDim.x) m = fmaxf(m, row[i]);
  m = waveMax(m);
  if (lane == 0) sred[wid] = m;
  __syncthreads();
  if (tid == 0) {
    float t = sred[0];
    for (int i = 1; i < nw; ++i) t = fmaxf(t, sred[i]);
    sred[0] = t;
  }
  __syncthreads();
  m = sred[0];
  __syncthreads();

  float s = 0.0f;
  for (int i = tid; i < cols; i += blockDim.x) s += expf(row[i] - m);
  s = waveSum(s);
  if (lane == 0) sred[wid] = s;
  __syncthreads();
  if (tid == 0) {
    float t = 0.0f;
    for (int i = 0; i < nw; ++i) t += sred[i];
    sred[0] = t;
  }
  __syncthreads();
  float lse = m + logf(sred[0]);
  for (int i = tid; i < cols; i += blockDim.x) row[i] -= lse;
}

// ---------------- host launch helpers ----------------
static inline int rowBlocks(int rows) { return (rows + WPB - 1) / WPB; }

static void gemm(hipStream_t st, const float* A, int lda, const float* B, int ldbK, int ldbN,
                 float* C, int ldc, const float* bias, const float* rs,
                 int M, int Nc, int K, float alpha, int acc) {
  int tiles = (M / 16) * (Nc / 16);
  int blocks = (tiles + WPB - 1) / WPB;
  k_wmma_gemm<<<blocks, 256, 0, st>>>(A, lda, B, ldbK, ldbN, C, ldc, bias, rs, M, Nc, K,
                                      alpha, acc);
}

static void fill(hipStream_t st, float* p, float v, size_t n) {
  int blocks = (int)((n + 255) / 256);
  if (blocks > 2048) blocks = 2048;
  k_fill<<<blocks, 256, 0, st>>>(p, v, n);
}

extern "C" void kernel_launch(void* const* d_in, const int* in_sizes, int n_in,
                              void* d_out, int out_size, void* d_ws, size_t ws_size,
                              hipStream_t stream) {
  (void)in_sizes; (void)n_in; (void)out_size; (void)ws_size;
  const int*   iid    = (const int*)d_in[0];
  const int*   src    = (const int*)d_in[1];
  const int*   dst    = (const int*)d_in[2];
  const float* edge_w = (const float*)d_in[3];
  const float* edge_t = (const float*)d_in[4];
  const float* node_t = (const float*)d_in[5];
  const float* emb    = (const float*)d_in[6];
  const float* W1     = (const float*)d_in[7];
  const float* W2     = (const float*)d_in[8];
  const float* gw_ih  = (const float*)d_in[9];   // (384,256)
  const float* gw_hh  = (const float*)d_in[10];  // (384,128)
  const float* gb_ih  = (const float*)d_in[11];
  const float* gb_hh  = (const float*)d_in[12];
  const float* Wxr = (const float*)d_in[13]; const float* bxr = (const float*)d_in[14];
  const float* Wxz = (const float*)d_in[15]; const float* bxz = (const float*)d_in[16];
  const float* Wxh = (const float*)d_in[17]; const float* bxh = (const float*)d_in[18];
  const float* Whr = (const float*)d_in[19]; const float* bhr = (const float*)d_in[20];
  const float* Whz = (const float*)d_in[21]; const float* bhz = (const float*)d_in[22];
  const float* Whh = (const float*)d_in[23]; const float* bhh = (const float*)d_in[24];
  const float* Wu  = (const float*)d_in[25];
  const float* Wv  = (const float*)d_in[26];
  const float* bv  = (const float*)d_in[27];
  const float* We  = (const float*)d_in[28];
  const float* Wsr = (const float*)d_in[29];   // (256,128)
  float* out = (float*)d_out;

  // -------- workspace layout (floats) --------
  float* W = (float*)d_ws;
  size_t off = 0;
  auto alloc = [&](size_t n) { float* p = W + off; off += n; return p; };
  float* target = alloc((size_t)(kV + LCOLS) * kD);  // l2norm(emb), +pad for tail block
  float* xbuf   = alloc((size_t)kN * kD);            // feat / x / fb
  float* hbuf   = alloc((size_t)kN * kD);            // h
  float* uni    = alloc((size_t)kN * 1024);          // union (GRU / ODE / attention)
  float* den1   = alloc(kN);
  float* den2   = alloc(kN);
  float* deg    = alloc(kN);
  float* dis    = alloc(kN);
  float* emf    = alloc(kE);
  float* tmax   = alloc(4);
  float* evec   = alloc(kN);

  const size_t ND = (size_t)kN * kD;
  // GRU-phase aliases
  float* hn   = uni;                        // N x 256
  float* num1 = uni + (size_t)kN * 256;     // N x 128  (dead after hn)
  float* num2 = uni + (size_t)kN * 384;     // N x 128  (dead after hn)
  float* gi   = uni + (size_t)kN * 256;     // N x 384  (overwrites num1/2)
  float* gh   = uni + (size_t)kN * 640;     // N x 384
  // ODE-phase aliases
  float* aggX  = uni;
  float* aggH  = uni + (size_t)kN * 128;
  float* aggRH = uni + (size_t)kN * 256;
  float* rbuf  = uni + (size_t)kN * 384;
  float* zbuf  = uni + (size_t)kN * 512;
  float* ubuf  = uni + (size_t)kN * 640;
  float* rhbuf = uni + (size_t)kN * 768;
  // attention-phase aliases
  float* uall  = uni;                       // N x 128
  float* lv    = uni + (size_t)kN * 128;    // 512 x 128
  float* srin  = uni + (size_t)kN * 256;    // 512 x 256
  float* srpre = uni + (size_t)kN * 384;    // 512 x 128

  // ---- 1. target = l2norm(emb); feat = target[iid] ----
  k_l2norm_rows<<<rowBlocks(kV), 256, 0, stream>>>(emb, target, kV);
  k_gather_rows<<<rowBlocks(kN), 256, 0, stream>>>(target, iid, xbuf, kN);

  // ---- 2. weighted-mean neighbor aggregation (both directions) ----
  fill(stream, num1, 0.0f, 2 * ND);           // num1,num2 contiguous
  fill(stream, den1, 0.0f, 2 * (size_t)kN);   // den1,den2 contiguous
  k_gru_edge_scatter<<<rowBlocks(kE), 256, 0, stream>>>(src, dst, edge_w, xbuf,
                                                        num1, num2, den1, den2);
  k_div_den<<<2048, 256, 0, stream>>>(num1, den1, ND);
  k_div_den<<<2048, 256, 0, stream>>>(num2, den2, ND);

  // ---- 3. hn = [neigh1@W1, neigh2@W2];  gi = hn @ gw_ih.T;  gh = feat @ gw_hh.T ----
  gemm(stream, num1, kD, W1, kD, 1, hn,        256, nullptr, nullptr, kN, kD, kD, 1.0f, 0);
  gemm(stream, num2, kD, W2, kD, 1, hn + 128,  256, nullptr, nullptr, kN, kD, kD, 1.0f, 0);
  gemm(stream, hn,   256, gw_ih, 1, 256, gi, 384, gb_ih, nullptr, kN, 384, 256, 1.0f, 0);
  gemm(stream, xbuf, kD,  gw_hh, 1, 128, gh, 384, gb_hh, nullptr, kN, 384, kD,  1.0f, 0);
  k_gru_gate_norm<<<rowBlocks(kN), 256, 0, stream>>>(gi, gh, xbuf, hbuf, kN);

  // ---- 4. t_end = max(edge_t) ----
  fill(stream, tmax, 0.0f, 1);
  k_reduce_max_pos<<<120, 256, 0, stream>>>(edge_t, kE, tmax);

  // ---- 5. ODE: 4 Euler steps ----
  for (int kstep = 0; kstep < kSplits; ++kstep) {
    fill(stream, aggX, 0.0f, 2 * ND);       // aggX,aggH contiguous
    fill(stream, deg, 0.0f, kN);
    k_ode_mask_deg<<<(kE + 255) / 256, 256, 0, stream>>>(src, dst, edge_t, node_t,
                                                         tmax, kstep, emf, deg);
    k_compute_dis<<<(kN + 255) / 256, 256, 0, stream>>>(deg, dis, kN);
    k_ode_scatter_xh<<<rowBlocks(kE), 256, 0, stream>>>(src, dst, emf, dis,
                                                        xbuf, hbuf, aggX, aggH);
    // rr pre-activation: (dis*aggX)@Wxr + bxr + (dis*aggH)@Whr + bhr
    gemm(stream, aggX, kD, Wxr, kD, 1, rbuf, kD, bxr, dis, kN, kD, kD, 1.0f, 0);
    gemm(stream, aggH, kD, Whr, kD, 1, rbuf, kD, bhr, dis, kN, kD, kD, 1.0f, 1);
    // zz pre-activation
    gemm(stream, aggX, kD, Wxz, kD, 1, zbuf, kD, bxz, dis, kN, kD, kD, 1.0f, 0);
    gemm(stream, aggH, kD, Whz, kD, 1, zbuf, kD, bhz, dis, kN, kD, kD, 1.0f, 1);
    // gates + rh = rr*h
    k_ode_gates_rz<<<2048, 256, 0, stream>>>(rbuf, zbuf, hbuf, rhbuf, ND);
    fill(stream, aggRH, 0.0f, ND);
    k_ode_scatter_one<<<rowBlocks(kE), 256, 0, stream>>>(src, dst, emf, dis, rhbuf, aggRH);
    // uu pre-activation
    gemm(stream, aggX,  kD, Wxh, kD, 1, ubuf, kD, bxh, dis, kN, kD, kD, 1.0f, 0);
    gemm(stream, aggRH, kD, Whh, kD, 1, ubuf, kD, bhh, dis, kN, kD, kD, 1.0f, 1);
    k_ode_update_h<<<2048, 256, 0, stream>>>(ubuf, zbuf, hbuf, node_t, tmax, kstep, ND);
  }

  // ---- 6. fb = l2norm(h) ----
  k_l2norm_rows<<<rowBlocks(kN), 256, 0, stream>>>(hbuf, xbuf, kN);

  // ---- 7. attention readout ----
  gemm(stream, xbuf, kD, Wu, kD, 1, uall, kD, nullptr, nullptr, kN, kD, kD, 1.0f, 0);
  gemm(stream, xbuf + (size_t)(kNPG - 1) * kD, kNPG * kD, Wv, kD, 1, lv, kD, bv, nullptr,
       kB, kD, kD, 1.0f, 0);
  k_attn_e<<<rowBlocks(kN), 256, 0, stream>>>(uall, lv, We, evec, kN);
  k_attn_combine<<<kB, kD, 0, stream>>>(xbuf, evec, srin);

  // ---- 8. sr = l2norm(srin @ W_sr) ----
  gemm(stream, srin, 256, Wsr, kD, 1, srpre, kD, nullptr, nullptr, kB, kD, 256, 1.0f, 0);
  k_l2norm_rows<<<rowBlocks(kB), 256, 0, stream>>>(srpre, srpre, kB);

  // ---- 9. logits = SCALE * sr @ target^T  (LDS-staged async B tiles) ----
  {
    dim3 lgrid((kV + LCOLS - 1) / LCOLS, kB / 16);   // 782 x 32
    k_logits_gemm<<<lgrid, 256, 0, stream>>>(srpre, target, out);
  }

  // ---- 10. row log-softmax over V ----
  k_log_softmax_rows<<<kB, 256, 0, stream>>>(out, kV);
}